// GraphSAGE_20925080666658
// MI455X (gfx1250) — compile-verified
//
#include <hip/hip_runtime.h>
#include <hip/hip_bf16.h>

typedef __attribute__((ext_vector_type(2))) float v2f;
typedef __attribute__((ext_vector_type(4))) float v4f;
typedef __attribute__((ext_vector_type(8))) float v8f;

#define D_F 128   // feature width of every matmul output
#define EPS 1e-5f

__device__ __forceinline__ v8f wmma_f32(v2f a, v2f b, v8f c) {
    // D = A(16x4,f32) * B(4x16,f32) + C(16x16,f32)
    return __builtin_amdgcn_wmma_f32_16x16x4_f32(
        /*neg_a=*/false, a, /*neg_b=*/false, b,
        /*c_mod=*/(short)0, c, /*reuse_a=*/false, /*reuse_b=*/false);
}

// ---------------------------------------------------------------------------
// Zero fill (agg + cnt must be zeroed every call; d_ws is poisoned 0xAA)
// ---------------------------------------------------------------------------
__global__ void fill_zero_kernel(float* __restrict__ p, long long n) {
    long long i = (long long)blockIdx.x * blockDim.x + threadIdx.x;
    long long stride = (long long)gridDim.x * blockDim.x;
    for (; i < n; i += stride) p[i] = 0.0f;
}

// ---------------------------------------------------------------------------
// out[M,128] = A[M,K] @ W[K,128] + bias        (node transform, K=128)
// Block: 256 threads = 8 waves; wave w owns output columns [16w,16w+16).
// ---------------------------------------------------------------------------
template <int K>
__global__ void gemm_bias_kernel(const float* __restrict__ A,
                                 const float* __restrict__ W,
                                 const float* __restrict__ bias,
                                 float* __restrict__ out, int M) {
    __shared__ float sA[16 * K];
    const int mbase = blockIdx.x * 16;
    const int wave  = threadIdx.x >> 5;
    const int lane  = threadIdx.x & 31;
    const int K4    = K / 4;                 // float4s per row

    if (mbase + 16 <= M) {                   // full tile: vectorized staging
        for (int i = threadIdx.x; i < 16 * K4; i += 256) {
            int r = i / K4, c4 = i % K4;
            v4f v = *(const v4f*)(A + (long long)(mbase + r) * K + c4 * 4);
            *(v4f*)(sA + r * K + c4 * 4) = v;
        }
    } else {
        for (int i = threadIdx.x; i < 16 * K; i += 256) {
            int r = i / K, c = i % K;
            int row = mbase + r;
            sA[i] = (row < M) ? A[(long long)row * K + c] : 0.0f;
        }
    }
    __syncthreads();

    const int fbase = wave * 16;
    const int half  = lane >> 4;   // 0: K pair {k,k+1}, 1: K pair {k+2,k+3}
    const int l16   = lane & 15;

    v8f c = {};
    #pragma unroll 8
    for (int k = 0; k < K; k += 4) {
        v2f a, b;
        a.x = sA[l16 * K + k + 2 * half];
        a.y = sA[l16 * K + k + 2 * half + 1];
        b.x = W[(k + 2 * half)     * D_F + fbase + l16];
        b.y = W[(k + 2 * half + 1) * D_F + fbase + l16];
        c = wmma_f32(a, b, c);
    }

    const float bcol = bias[fbase + l16];
    if (mbase + 16 <= M) {
        #pragma unroll
        for (int r = 0; r < 8; ++r)
            out[(long long)(mbase + r + 8 * half) * D_F + fbase + l16] = c[r] + bcol;
    } else {
        #pragma unroll
        for (int r = 0; r < 8; ++r) {
            int m = mbase + r + 8 * half;
            if (m < M) out[(long long)m * D_F + fbase + l16] = c[r] + bcol;
        }
    }
}

// ---------------------------------------------------------------------------
// Fused message + scatter: per 16-edge tile,
//   C = node_t[src] (gather) + be, then C += edge_attr(16x32) @ We(32x128),
//   then atomic scatter-add of each C row into agg[dst], and cnt[dst] += 1.
// ---------------------------------------------------------------------------
__global__ void edge_msg_scatter_kernel(const float* __restrict__ node_t,  // [N,128]
                                        const long long* __restrict__ src,
                                        const long long* __restrict__ dst,
                                        const float* __restrict__ edge_attr, // [E,32]
                                        const float* __restrict__ We,        // [32,128]
                                        const float* __restrict__ be,
                                        float* __restrict__ agg,             // [N,128]
                                        float* __restrict__ cnt,             // [N]
                                        int E) {
    __shared__ float sEA[16 * 32];
    __shared__ int   sSrc[16];
    __shared__ int   sDst[16];

    const int  ebase = blockIdx.x * 16;
    const bool full  = (ebase + 16 <= E);
    const int  wave  = threadIdx.x >> 5;
    const int  lane  = threadIdx.x & 31;

    if (threadIdx.x < 16) {
        int e = ebase + threadIdx.x;
        if (e < E) {
            sSrc[threadIdx.x] = (int)src[e];
            sDst[threadIdx.x] = (int)dst[e];
        } else {
            sSrc[threadIdx.x] = 0;
            sDst[threadIdx.x] = -1;
        }
    }
    if (full) {                      // 16 edges * 32 feats = 128 float4
        if (threadIdx.x < 128) {
            v4f v = *(const v4f*)(edge_attr + (long long)ebase * 32 + threadIdx.x * 4);
            *(v4f*)(sEA + threadIdx.x * 4) = v;
        }
    } else {
        for (int i = threadIdx.x; i < 16 * 32; i += 256) {
            long long idx = (long long)ebase * 32 + i;
            sEA[i] = (idx < (long long)E * 32) ? edge_attr[idx] : 0.0f;
        }
    }
    __syncthreads();

    const int fbase = wave * 16;
    const int half  = lane >> 4;
    const int l16   = lane & 15;
    const int col   = fbase + l16;
    const float bcol = be[col];

    // Hoist indices into registers (prevents LDS-aliasing reloads around atomics)
    int srcR[8], dstR[8];
    #pragma unroll
    for (int r = 0; r < 8; ++r) {
        int m = r + 8 * half;
        srcR[r] = sSrc[m];
        dstR[r] = sDst[m];
    }

    v8f c;
    #pragma unroll
    for (int r = 0; r < 8; ++r)      // init C with gathered node transform + edge bias
        c[r] = node_t[(long long)srcR[r] * D_F + col] + bcol;

    #pragma unroll
    for (int k = 0; k < 32; k += 4) {
        v2f a, b;
        a.x = sEA[l16 * 32 + k + 2 * half];
        a.y = sEA[l16 * 32 + k + 2 * half + 1];
        b.x = We[(k + 2 * half)     * D_F + col];
        b.y = We[(k + 2 * half + 1) * D_F + col];
        c = wmma_f32(a, b, c);
    }

    if (full) {
        #pragma unroll
        for (int r = 0; r < 8; ++r)
            atomicAdd(&agg[(long long)dstR[r] * D_F + col], c[r]);
    } else {
        #pragma unroll
        for (int r = 0; r < 8; ++r)
            if (dstR[r] >= 0) atomicAdd(&agg[(long long)dstR[r] * D_F + col], c[r]);
    }
    if (wave == 0 && lane < 16) {
        int d = sDst[lane];
        if (d >= 0) atomicAdd(&cnt[d], 1.0f);
    }
}

// ---------------------------------------------------------------------------
// Update: out = relu( concat(X, agg/max(cnt,1)) @ Wu + bu )   (K = 256)
// ---------------------------------------------------------------------------
__global__ void update_gemm_kernel(const float* __restrict__ X,    // [M,128]
                                   const float* __restrict__ agg,  // [M,128]
                                   const float* __restrict__ cnt,  // [M]
                                   const float* __restrict__ Wu,   // [256,128]
                                   const float* __restrict__ bu,
                                   float* __restrict__ out, int M) {
    __shared__ float sA[16 * 256];
    __shared__ float sInv[16];
    const int mbase = blockIdx.x * 16;
    const int wave  = threadIdx.x >> 5;
    const int lane  = threadIdx.x & 31;

    if (threadIdx.x < 16) {
        int row = mbase + threadIdx.x;
        float cn = (row < M) ? cnt[row] : 1.0f;
        sInv[threadIdx.x] = 1.0f / fmaxf(cn, 1.0f);
    }
    __syncthreads();

    if (mbase + 16 <= M) {           // full tile: vectorized staging, 512 float4 x2
        for (int i = threadIdx.x; i < 512; i += 256) {
            int r = i >> 5, c4 = i & 31;               // 32 float4 per 128-col row
            float inv = sInv[r];
            v4f vx = *(const v4f*)(X   + (long long)(mbase + r) * D_F + c4 * 4);
            v4f va = *(const v4f*)(agg + (long long)(mbase + r) * D_F + c4 * 4);
            va.x *= inv; va.y *= inv; va.z *= inv; va.w *= inv;
            *(v4f*)(sA + r * 256 + c4 * 4)       = vx;
            *(v4f*)(sA + r * 256 + 128 + c4 * 4) = va;
        }
    } else {
        for (int i = threadIdx.x; i < 16 * 128; i += 256) {
            int r = i >> 7, col = i & 127;
            int row = mbase + r;
            if (row < M) {
                sA[r * 256 + col]       = X[(long long)row * D_F + col];
                sA[r * 256 + 128 + col] = agg[(long long)row * D_F + col] * sInv[r];
            } else {
                sA[r * 256 + col] = 0.0f;
                sA[r * 256 + 128 + col] = 0.0f;
            }
        }
    }
    __syncthreads();

    const int fbase = wave * 16;
    const int half  = lane >> 4;
    const int l16   = lane & 15;

    v8f c = {};
    #pragma unroll 8
    for (int k = 0; k < 256; k += 4) {
        v2f a, b;
        a.x = sA[l16 * 256 + k + 2 * half];
        a.y = sA[l16 * 256 + k + 2 * half + 1];
        b.x = Wu[(k + 2 * half)     * D_F + fbase + l16];
        b.y = Wu[(k + 2 * half + 1) * D_F + fbase + l16];
        c = wmma_f32(a, b, c);
    }

    const float bcol = bu[fbase + l16];
    if (mbase + 16 <= M) {
        #pragma unroll
        for (int r = 0; r < 8; ++r)
            out[(long long)(mbase + r + 8 * half) * D_F + fbase + l16] =
                fmaxf(c[r] + bcol, 0.0f);
    } else {
        #pragma unroll
        for (int r = 0; r < 8; ++r) {
            int m = mbase + r + 8 * half;
            if (m < M) out[(long long)m * D_F + fbase + l16] = fmaxf(c[r] + bcol, 0.0f);
        }
    }
}

// ---------------------------------------------------------------------------
// Row LayerNorm over 128 features; one wave per row, 4 floats per lane.
// ---------------------------------------------------------------------------
__global__ void layernorm_kernel(const float* __restrict__ in,
                                 const float* __restrict__ g,
                                 const float* __restrict__ beta,
                                 float* __restrict__ out, int M) {
    const int row  = blockIdx.x * 8 + (threadIdx.x >> 5);
    const int lane = threadIdx.x & 31;
    if (row >= M) return;

    const float* p = in + (long long)row * D_F;
    v4f v  = *(const v4f*)(p + lane * 4);
    float s  = v.x + v.y + v.z + v.w;
    float s2 = v.x * v.x + v.y * v.y + v.z * v.z + v.w * v.w;
    #pragma unroll
    for (int off = 16; off > 0; off >>= 1) {
        s  += __shfl_xor(s,  off, 32);
        s2 += __shfl_xor(s2, off, 32);
    }
    const float mu  = s * (1.0f / D_F);
    const float var = s2 * (1.0f / D_F) - mu * mu;
    const float inv = rsqrtf(var + EPS);

    v4f gv = *(const v4f*)(g + lane * 4);
    v4f bv = *(const v4f*)(beta + lane * 4);
    v4f o;
    o.x = (v.x - mu) * inv * gv.x + bv.x;
    o.y = (v.y - mu) * inv * gv.y + bv.y;
    o.z = (v.z - mu) * inv * gv.z + bv.z;
    o.w = (v.w - mu) * inv * gv.w + bv.w;
    *(v4f*)(out + (long long)row * D_F + lane * 4) = o;
}

// ---------------------------------------------------------------------------
extern "C" void kernel_launch(void* const* d_in, const int* in_sizes, int n_in,
                              void* d_out, int out_size, void* d_ws, size_t ws_size,
                              hipStream_t stream) {
    const float*     x    = (const float*)d_in[0];
    const long long* eidx = (const long long*)d_in[1];   // int64 per reference
    const float*     ea   = (const float*)d_in[2];
    const float *Wn1 = (const float*)d_in[3],  *bn1 = (const float*)d_in[4];
    const float *We1 = (const float*)d_in[5],  *be1 = (const float*)d_in[6];
    const float *Wu1 = (const float*)d_in[7],  *bu1 = (const float*)d_in[8];
    const float *g1  = (const float*)d_in[9],  *beta1 = (const float*)d_in[10];
    const float *Wn2 = (const float*)d_in[11], *bn2 = (const float*)d_in[12];
    const float *We2 = (const float*)d_in[13], *be2 = (const float*)d_in[14];
    const float *Wu2 = (const float*)d_in[15], *bu2 = (const float*)d_in[16];
    const float *g2  = (const float*)d_in[17], *beta2 = (const float*)d_in[18];

    const int N = in_sizes[0] / D_F;
    const int E = in_sizes[2] / 32;
    const long long* src = eidx;
    const long long* dst = eidx + E;

    float* B0 = (float*)d_ws;                    // node_t / pre-LN scratch [N,128]
    float* B1 = B0 + (size_t)N * D_F;            // agg [N,128]
    float* B2 = B1 + (size_t)N * D_F;            // cnt [N]  (contiguous after agg)
    float* B3 = B2 + (size_t)N;                  // h1 post-LN [N,128]
    float* out = (float*)d_out;

    const dim3 blk(256);
    const int mtiles = (N + 15) / 16;
    const int etiles = (E + 15) / 16;
    const int lnblocks = (N + 7) / 8;

    // ---- layer 1 ----
    fill_zero_kernel<<<2048, blk, 0, stream>>>(B1, (long long)N * D_F + N);
    gemm_bias_kernel<128><<<mtiles, blk, 0, stream>>>(x, Wn1, bn1, B0, N);
    edge_msg_scatter_kernel<<<etiles, blk, 0, stream>>>(B0, src, dst, ea, We1, be1, B1, B2, E);
    update_gemm_kernel<<<mtiles, blk, 0, stream>>>(x, B1, B2, Wu1, bu1, B0, N);
    layernorm_kernel<<<lnblocks, blk, 0, stream>>>(B0, g1, beta1, B3, N);

    // ---- layer 2 ----
    fill_zero_kernel<<<2048, blk, 0, stream>>>(B1, (long long)N * D_F + N);
    gemm_bias_kernel<128><<<mtiles, blk, 0, stream>>>(B3, Wn2, bn2, B0, N);
    edge_msg_scatter_kernel<<<etiles, blk, 0, stream>>>(B0, src, dst, ea, We2, be2, B1, B2, E);
    update_gemm_kernel<<<mtiles, blk, 0, stream>>>(B3, B1, B2, Wu2, bu2, out, N);
    layernorm_kernel<<<lnblocks, blk, 0, stream>>>(out, g2, beta2, out, N);
}